// Net_MP_Gauss_57775900066585
// MI455X (gfx1250) — compile-verified
//
#include <hip/hip_runtime.h>
#include <hip/hip_bf16.h>

// GNN NNConv, 4 shared-weight message-passing steps, W=32.
// Strategy: factor the per-edge 32x32 matrix through the source node:
//   P[n, c*32+o] = sum_i h[n,i] * k2_w[i*32+o, c]   (c<8),  P[n,256+o] = sum_i h[n,i]*k2_b[i*32+o]
//   msg[e,o]     = sum_{c<9} ehc[e,c] * P[src[e], c*32+o]   with ehc[:,8]==1
// P is 18.9 MB -> L2 resident on MI455X (192 MB L2). GEMMs use V_WMMA_F32_16X16X4_F32.

typedef __attribute__((ext_vector_type(2))) float v2f;
typedef __attribute__((ext_vector_type(8))) float v8f;

#define WIDTH 32
#define PCOLS 288   // 8*32 + 32 (bias column block)

// ---------- small setup kernels ----------

// Bmat[i*288 + c*32 + o] = k2_w[(i*32+o)*8 + c] (c<8);  Bmat[i*288+256+o] = k2_b[i*32+o]
__global__ void k_build_bmat(const float* __restrict__ k2w, const float* __restrict__ k2b,
                             float* __restrict__ Bmat) {
    int idx = blockIdx.x * blockDim.x + threadIdx.x;   // 32*288
    if (idx >= WIDTH * PCOLS) return;
    int i = idx / PCOLS, t = idx % PCOLS;
    float v;
    if (t < 256) {
        int c = t >> 5, o = t & 31;
        v = k2w[(i * 32 + o) * 8 + c];
    } else {
        int o = t - 256;
        v = k2b[i * 32 + o];
    }
    Bmat[idx] = v;
}

// h[n,o] = x[n,:3] . fc1_w[o,:3] + fc1_b[o]
__global__ void k_encode(const float* __restrict__ x, const float* __restrict__ w,
                         const float* __restrict__ b, float* __restrict__ h, int n_nodes) {
    int idx = blockIdx.x * blockDim.x + threadIdx.x;   // N*32
    if (idx >= n_nodes * WIDTH) return;
    int n = idx >> 5, o = idx & 31;
    float a = b[o];
    #pragma unroll
    for (int c = 0; c < 3; ++c) a += x[n * 3 + c] * w[o * 3 + c];
    h[idx] = a;
}

// ehc[e,0..7] = relu(edge_attr[e,:7] . k1_w[c,:7] + k1_b[c]);  ehc[e,8] = 1
__global__ void k_edge_feat(const float* __restrict__ ea, const float* __restrict__ w,
                            const float* __restrict__ b, float* __restrict__ ehc, int n_edges) {
    int e = blockIdx.x * blockDim.x + threadIdx.x;
    if (e >= n_edges) return;
    float a[7];
    #pragma unroll
    for (int j = 0; j < 7; ++j) a[j] = ea[e * 7 + j];
    #pragma unroll
    for (int c = 0; c < 8; ++c) {
        float s = b[c];
        #pragma unroll
        for (int j = 0; j < 7; ++j) s += a[j] * w[c * 7 + j];
        ehc[e * 9 + c] = fmaxf(s, 0.0f);
    }
    ehc[e * 9 + 8] = 1.0f;
}

__global__ void k_zero(float* __restrict__ p, int n) {
    int i = blockIdx.x * blockDim.x + threadIdx.x;
    if (i < n) p[i] = 0.0f;
}

__global__ void k_deg_count(const int* __restrict__ dst, float* __restrict__ deg, int n_edges) {
    int e = blockIdx.x * blockDim.x + threadIdx.x;
    if (e < n_edges) atomicAdd(&deg[dst[e]], 1.0f);
}

__global__ void k_inv_deg(const float* __restrict__ deg, float* __restrict__ invdeg, int n_nodes) {
    int n = blockIdx.x * blockDim.x + threadIdx.x;
    if (n < n_nodes) invdeg[n] = 1.0f / fmaxf(deg[n], 1.0f);
}

// ---------- WMMA GEMM: C[M,ncols] = A[M,32] * B[32,ncols] (+ bias[col]) ----------
// One wave computes one 16x16 tile using V_WMMA_F32_16X16X4_F32 (K=32 -> 8 steps).
// A fragment (16x4): lanes 0-15 hold M=lane, K={k,k+1}; lanes 16-31 hold M=lane-16, K={k+2,k+3}.
// B fragment (4x16): lanes 0-15 hold N=lane, K rows {k,k+1}; lanes 16-31 hold K rows {k+2,k+3}.
// C/D: VGPR r -> row (half*8 + r), col = lane&15 (ISA 7.12.2).
__global__ void k_gemm32_wmma(const float* __restrict__ A, const float* __restrict__ B,
                              const float* __restrict__ bias, float* __restrict__ C,
                              int M, int ncols) {
    const int wavesPerBlock = blockDim.x >> 5;
    const int wid  = blockIdx.x * wavesPerBlock + (threadIdx.x >> 5);
    const int ntiles = ncols >> 4;
    const int mt = wid / ntiles;
    const int nt = wid % ntiles;
    if (mt * 16 >= M) return;
    const int lane = threadIdx.x & 31;
    const int half = lane >> 4;
    const int l15  = lane & 15;

    const float* arow = A + (size_t)(mt * 16 + l15) * WIDTH;
    const float* bcol = B + nt * 16 + l15;

    v8f acc = {};
    #pragma unroll
    for (int k = 0; k < WIDTH; k += 4) {
        v2f a, b;
        a.x = arow[k + 2 * half + 0];
        a.y = arow[k + 2 * half + 1];
        b.x = bcol[(size_t)(k + 2 * half + 0) * ncols];
        b.y = bcol[(size_t)(k + 2 * half + 1) * ncols];
        acc = __builtin_amdgcn_wmma_f32_16x16x4_f32(false, a, false, b,
                                                    (short)0, acc, false, false);
    }

    const int col = nt * 16 + l15;
    const float bv = bias ? bias[col] : 0.0f;
    #pragma unroll
    for (int r = 0; r < 8; ++r) {
        int row = mt * 16 + half * 8 + r;
        C[(size_t)row * ncols + col] = acc[r] + bv;
    }
}

// ---------- per-edge message + scatter-mean ----------
// wave per edge; lane o: msg_o = sum_c ehc[e,c]*P[src, c*32+o]; atomicAdd(agg[dst,o], msg_o/deg[dst])
__global__ void k_edge_msg(const int* __restrict__ src, const int* __restrict__ dst,
                           const float* __restrict__ ehc, const float* __restrict__ P,
                           const float* __restrict__ invdeg, float* __restrict__ agg,
                           int n_edges) {
    const int e = blockIdx.x * (blockDim.x >> 5) + (threadIdx.x >> 5);
    if (e >= n_edges) return;
    const int lane = threadIdx.x & 31;
    const int s = src[e], d = dst[e];
    const float* __restrict__ prow = P + (size_t)s * PCOLS;
    const float* __restrict__ ec   = ehc + (size_t)e * 9;
    float m = 0.0f;
    #pragma unroll
    for (int c = 0; c < 9; ++c) m += ec[c] * prow[c * 32 + lane];
    atomicAdd(&agg[(size_t)d * WIDTH + lane], m * invdeg[d]);
}

__global__ void k_relu_copy(const float* __restrict__ a, float* __restrict__ h, int n) {
    int i = blockIdx.x * blockDim.x + threadIdx.x;
    if (i < n) h[i] = fmaxf(a[i], 0.0f);
}

// out[n] = h[n,:] . fc2_w + fc2_b
__global__ void k_head(const float* __restrict__ h, const float* __restrict__ w,
                       const float* __restrict__ b, float* __restrict__ out, int n_nodes) {
    int n = blockIdx.x * blockDim.x + threadIdx.x;
    if (n >= n_nodes) return;
    float a = b[0];
    #pragma unroll
    for (int o = 0; o < WIDTH; ++o) a += h[(size_t)n * WIDTH + o] * w[o];
    out[n] = a;
}

extern "C" void kernel_launch(void* const* d_in, const int* in_sizes, int n_in,
                              void* d_out, int out_size, void* d_ws, size_t ws_size,
                              hipStream_t stream) {
    const float* x      = (const float*)d_in[0];
    const int*   ei     = (const int*)  d_in[1];
    const float* eattr  = (const float*)d_in[2];
    const float* fc1_w  = (const float*)d_in[3];
    const float* fc1_b  = (const float*)d_in[4];
    const float* k1_w   = (const float*)d_in[5];
    const float* k1_b   = (const float*)d_in[6];
    const float* k2_w   = (const float*)d_in[7];
    const float* k2_b   = (const float*)d_in[8];
    const float* root_w = (const float*)d_in[9];
    const float* conv_b = (const float*)d_in[10];
    const float* fc2_w  = (const float*)d_in[11];
    const float* fc2_b  = (const float*)d_in[12];
    float* out = (float*)d_out;

    const int N = in_sizes[0] / 3;      // 16384
    const int E = in_sizes[1] / 2;      // 131072
    const int* src = ei;
    const int* dst = ei + E;

    // workspace partition (floats)
    float* ws     = (float*)d_ws;
    float* h      = ws;                       // N*32
    float* agg    = h      + (size_t)N * 32;  // N*32
    float* P      = agg    + (size_t)N * 32;  // N*288
    float* ehc    = P      + (size_t)N * PCOLS; // E*9
    float* invdeg = ehc    + (size_t)E * 9;   // N
    float* Bmat   = invdeg + (size_t)N;       // 32*288
    float* deg    = Bmat   + (size_t)WIDTH * PCOLS; // N

    const int T = 256;

    // one-time setup
    k_build_bmat<<<(WIDTH * PCOLS + T - 1) / T, T, 0, stream>>>(k2_w, k2_b, Bmat);
    k_encode   <<<(N * 32 + T - 1) / T, T, 0, stream>>>(x, fc1_w, fc1_b, h, N);
    k_edge_feat<<<(E + T - 1) / T, T, 0, stream>>>(eattr, k1_w, k1_b, ehc, E);
    k_zero     <<<(N + T - 1) / T, T, 0, stream>>>(deg, N);
    k_deg_count<<<(E + T - 1) / T, T, 0, stream>>>(dst, deg, E);
    k_inv_deg  <<<(N + T - 1) / T, T, 0, stream>>>(deg, invdeg, N);

    const int wavesPerBlock = T / 32;
    const int pTiles   = (N / 16) * (PCOLS / 16);  // 1024*18
    const int rTiles   = (N / 16) * (WIDTH / 16);  // 1024*2
    const int pBlocks  = (pTiles + wavesPerBlock - 1) / wavesPerBlock;
    const int rBlocks  = (rTiles + wavesPerBlock - 1) / wavesPerBlock;
    const int eBlocks  = (E + wavesPerBlock - 1) / wavesPerBlock;

    for (int depth = 0; depth < 4; ++depth) {
        // P = h @ Bmat          [N,32]x[32,288]  (WMMA)
        k_gemm32_wmma<<<pBlocks, T, 0, stream>>>(h, Bmat, nullptr, P, N, PCOLS);
        // agg = h @ root_w + conv_bias   [N,32]x[32,32]  (WMMA)
        k_gemm32_wmma<<<rBlocks, T, 0, stream>>>(h, root_w, conv_b, agg, N, WIDTH);
        // scatter-mean of per-edge messages into agg
        k_edge_msg<<<eBlocks, T, 0, stream>>>(src, dst, ehc, P, invdeg, agg, E);
        // h = relu(agg)
        k_relu_copy<<<(N * 32 + T - 1) / T, T, 0, stream>>>(agg, h, N * 32);
    }

    k_head<<<(N + T - 1) / T, T, 0, stream>>>(h, fc2_w, fc2_b, out, N);
}